// ReEig_16063177687356
// MI455X (gfx1250) — compile-verified
//
#include <hip/hip_runtime.h>
#include <math.h>

// CDNA5 WMMA vector operand types (wave32): f32 16x16x4 -> A/B = 2 VGPRs, C/D = 8 VGPRs
typedef __attribute__((ext_vector_type(2))) float v2f;
typedef __attribute__((ext_vector_type(8))) float v8f;

#define N 64
#define LDSP 65          // padded LDS stride (bank-conflict free)
#define NSWEEP 10        // fixed Jacobi sweep count (deterministic)
#define REEIG_EPS 1e-4f

#define XS(c, r) Xs[(c) * LDSP + (r)]
#define VS(c, r) Vs[(c) * LDSP + (r)]

__global__ __launch_bounds__(1024) void reeig_jacobi_kernel(const float* __restrict__ x,
                                                            float* __restrict__ out) {
  __shared__ float Xs[N * LDSP];   // columns of X (later U*max(S,eps)), col-major
  __shared__ float Vs[N * LDSP];   // accumulated right rotations V, col-major
  __shared__ float dscale[N];

  const int tid  = threadIdx.x;
  const int lane = tid & 31;
  const int wave = tid >> 5;
  const long long base = (long long)blockIdx.x * (N * N);

  // ---- load X column-major (coalesced global reads), V = I ----
  for (int i = tid; i < N * N; i += 1024) {
    int r = i >> 6, c = i & 63;          // i = r*64 + c
    XS(c, r) = x[base + i];
    VS(c, r) = (r == c) ? 1.0f : 0.0f;
  }
  __syncthreads();

  // ---- one-sided Jacobi: orthogonalize columns of X, accumulate V ----
  for (int sweep = 0; sweep < NSWEEP; ++sweep) {
    for (int round = 0; round < 63; ++round) {
      // tournament pairing: 32 disjoint pairs per round, all C(64,2) per sweep
      int p, q;
      if (wave == 0) { p = 63; q = round; }
      else { p = (round + wave) % 63; q = (round - wave + 63) % 63; }

      float xp0 = XS(p, lane),      xq0 = XS(q, lane);
      float xp1 = XS(p, lane + 32), xq1 = XS(q, lane + 32);

      float app = xp0 * xp0 + xp1 * xp1;
      float aqq = xq0 * xq0 + xq1 * xq1;
      float apq = xp0 * xq0 + xp1 * xq1;
      // wave32 all-reduce
      for (int m = 16; m >= 1; m >>= 1) {
        app += __shfl_xor(app, m, 32);
        aqq += __shfl_xor(aqq, m, 32);
        apq += __shfl_xor(apq, m, 32);
      }

      if (apq * apq > 1e-14f * app * aqq) {   // wave-uniform branch
        float tau = (aqq - app) / (2.0f * apq);
        float t   = copysignf(1.0f / (fabsf(tau) + sqrtf(1.0f + tau * tau)), tau);
        float cc  = rsqrtf(1.0f + t * t);
        float ss  = cc * t;
        XS(p, lane)      = cc * xp0 - ss * xq0;
        XS(q, lane)      = ss * xp0 + cc * xq0;
        XS(p, lane + 32) = cc * xp1 - ss * xq1;
        XS(q, lane + 32) = ss * xp1 + cc * xq1;
        float vp0 = VS(p, lane),      vq0 = VS(q, lane);
        float vp1 = VS(p, lane + 32), vq1 = VS(q, lane + 32);
        VS(p, lane)      = cc * vp0 - ss * vq0;
        VS(q, lane)      = ss * vp0 + cc * vq0;
        VS(p, lane + 32) = cc * vp1 - ss * vq1;
        VS(q, lane + 32) = ss * vp1 + cc * vq1;
      }
      __syncthreads();
    }
  }

  // ---- singular values = column norms; rectification scale max(s,eps)/s ----
  if (tid < N) {
    float sum = 0.0f;
    for (int r = 0; r < N; ++r) { float v = XS(tid, r); sum += v * v; }
    float s = sqrtf(sum);
    dscale[tid] = fmaxf(s, REEIG_EPS) / fmaxf(s, 1e-30f);
  }
  __syncthreads();

  // A = U * max(S,eps): scale column i of X by dscale[i]
  for (int i = tid; i < N * N; i += 1024) {
    int c = i >> 6, r = i & 63;
    XS(c, r) *= dscale[c];
  }
  __syncthreads();

  // ---- out = A * V^T via V_WMMA_F32_16X16X4_F32 ----
  // A[m][k] = XS(k, m); B[k][n] = (V^T)[k][n] = V[n][k] = VS(k, n)
  // 16 waves x one 16x16 tile, 16 K-steps of 4.
  if (wave < 16) {
    const int tm = (wave >> 2) * 16;
    const int tn = (wave & 3) * 16;
    const int hl = lane >> 4;        // 0: lanes 0-15 (K=0,1), 1: lanes 16-31 (K=2,3)
    const int l  = lane & 15;
    v8f acc = {};
    for (int k0 = 0; k0 < N; k0 += 4) {
      v2f a, b;
      a.x = XS(k0 + 2 * hl,     tm + l);
      a.y = XS(k0 + 2 * hl + 1, tm + l);
      b.x = VS(k0 + 2 * hl,     tn + l);
      b.y = VS(k0 + 2 * hl + 1, tn + l);
      acc = __builtin_amdgcn_wmma_f32_16x16x4_f32(false, a, false, b,
                                                  (short)0, acc, false, false);
    }
    // C/D layout: VGPR j -> row j (lanes 0-15) / row j+8 (lanes 16-31), col = lane%16
    float* o = out + base;
#pragma unroll
    for (int j = 0; j < 8; ++j) {
      o[(tm + hl * 8 + j) * N + tn + l] = acc[j];
    }
  }
}

extern "C" void kernel_launch(void* const* d_in, const int* in_sizes, int n_in,
                              void* d_out, int out_size, void* d_ws, size_t ws_size,
                              hipStream_t stream) {
  (void)n_in; (void)d_ws; (void)ws_size; (void)out_size;
  const float* x = (const float*)d_in[0];
  float* out = (float*)d_out;
  const int batches = in_sizes[0] / (N * N);   // 8192
  reeig_jacobi_kernel<<<batches, 1024, 0, stream>>>(x, out);
}